// UVTransformer_61795989455185
// MI455X (gfx1250) — compile-verified
//
#include <hip/hip_runtime.h>

typedef __attribute__((ext_vector_type(2))) float v2f;
typedef __attribute__((ext_vector_type(4))) float v4f;
typedef __attribute__((ext_vector_type(8))) float v8f;

// ---------------------------------------------------------------------------
// Kernel 1: per-batch composite.  One wave32 per batch.
//   M  = R1 @ R0^T            (computed with V_WMMA_F32_16X16X4_F32)
//   A  = K @ M                 (3x3)
//   bv = K @ t1 - A @ t0       (3)
// comp[b*16 + 0..8] = A row-major, comp[b*16 + 9..11] = bv
// ---------------------------------------------------------------------------
__global__ __launch_bounds__(32)
void uvt_composite_kernel(const float* __restrict__ R0, const float* __restrict__ t0,
                          const float* __restrict__ R1, const float* __restrict__ t1,
                          const float* __restrict__ Km, float* __restrict__ comp) {
  const int b    = blockIdx.x;
  const int lane = threadIdx.x;      // wave32: one full wave, EXEC all ones
  const int lo   = lane & 15;
  const int hi   = lane >> 4;
  const int k0   = 2 * hi;
  const int k1   = 2 * hi + 1;

  const float* r0 = R0 + b * 9;
  const float* r1 = R1 + b * 9;

  // A-matrix (16x4 f32): lanes 0-15 hold K=0 (v0), K=1 (v1); lanes 16-31 hold K=2,3.
  // A[m][k] = R1[m][k] zero-padded beyond 3x3.
  v2f av;
  av.x = (lo < 3 && k0 < 3) ? r1[lo * 3 + k0] : 0.0f;
  av.y = (lo < 3 && k1 < 3) ? r1[lo * 3 + k1] : 0.0f;
  // B-matrix (4x16 f32), row-striped across lanes: v0 = rows K=0 (lanes 0-15) /
  // K=2 (lanes 16-31); v1 = rows K=1 / K=3.  B[k][n] = R0^T[k][n] = R0[n][k].
  v2f bm;
  bm.x = (lo < 3 && k0 < 3) ? r0[lo * 3 + k0] : 0.0f;
  bm.y = (lo < 3 && k1 < 3) ? r0[lo * 3 + k1] : 0.0f;

  v8f c = {};
  // D[m][n] = sum_k R1[m][k] * R0[n][k]  ->  M = R1 @ R0^T in top-left 3x3
  c = __builtin_amdgcn_wmma_f32_16x16x4_f32(false, av, false, bm,
                                            (short)0, c, false, false);

  // C/D layout: lane n (0-15), VGPR r -> D[m=r][n].  M[i][j] lives at lane j, c[i].
  const float M00 = __shfl(c[0], 0, 32), M01 = __shfl(c[0], 1, 32), M02 = __shfl(c[0], 2, 32);
  const float M10 = __shfl(c[1], 0, 32), M11 = __shfl(c[1], 1, 32), M12 = __shfl(c[1], 2, 32);
  const float M20 = __shfl(c[2], 0, 32), M21 = __shfl(c[2], 1, 32), M22 = __shfl(c[2], 2, 32);

  if (lane == 0) {
    const float Mm[3][3] = {{M00, M01, M02}, {M10, M11, M12}, {M20, M21, M22}};
    const float t0l[3] = {t0[b * 3 + 0], t0[b * 3 + 1], t0[b * 3 + 2]};
    const float t1l[3] = {t1[b * 3 + 0], t1[b * 3 + 1], t1[b * 3 + 2]};
    float* cp = comp + b * 16;
    float A[3][3];
#pragma unroll
    for (int i = 0; i < 3; ++i) {
#pragma unroll
      for (int j = 0; j < 3; ++j) {
        A[i][j] = Km[i * 3 + 0] * Mm[0][j] + Km[i * 3 + 1] * Mm[1][j] + Km[i * 3 + 2] * Mm[2][j];
        cp[i * 3 + j] = A[i][j];
      }
    }
#pragma unroll
    for (int i = 0; i < 3; ++i) {
      const float kt = Km[i * 3 + 0] * t1l[0] + Km[i * 3 + 1] * t1l[1] + Km[i * 3 + 2] * t1l[2];
      const float at = A[i][0] * t0l[0] + A[i][1] * t0l[1] + A[i][2] * t0l[2];
      cp[9 + i] = kt - at;
    }
  }
}

// ---------------------------------------------------------------------------
// Kernel 2: bandwidth-bound main pass.  4 pixels per thread, 128-bit accesses,
// non-temporal hints on streaming depth reads + all stores (keeps the 3 MB ray
// array resident in L2 across the 64 batch passes).
// ---------------------------------------------------------------------------
__global__ __launch_bounds__(256)
void uvt_main_kernel(const float* __restrict__ depth, const float* __restrict__ ray,
                     const float* __restrict__ comp, float* __restrict__ out_uv,
                     float* __restrict__ out_d, int npix) {
  const int b  = blockIdx.y;
  const int n0 = (blockIdx.x * 256 + threadIdx.x) * 4;
  if (n0 >= npix) return;

  const float* cp = comp + b * 16;
  const float a00 = cp[0], a01 = cp[1], a02 = cp[2];
  const float a10 = cp[3], a11 = cp[4], a12 = cp[5];
  const float a20 = cp[6], a21 = cp[7], a22 = cp[8];
  const float b0  = cp[9], b1  = cp[10], b2 = cp[11];

  const size_t base = (size_t)b * (size_t)npix + (size_t)n0;

  const v4f dep = __builtin_nontemporal_load((const v4f*)(depth + base));
  const v4f* rp = (const v4f*)(ray + (size_t)n0 * 3);  // 48B-aligned (n0 % 4 == 0)
  const v4f ra = rp[0], rb = rp[1], rc = rp[2];

  const float rr[12] = {ra.x, ra.y, ra.z, ra.w, rb.x, rb.y, rb.z, rb.w,
                        rc.x, rc.y, rc.z, rc.w};
  const float dd[4] = {dep.x, dep.y, dep.z, dep.w};

  float uv[8], dv[4];
#pragma unroll
  for (int p = 0; p < 4; ++p) {
    const float rx = rr[3 * p + 0], ry = rr[3 * p + 1], rz = rr[3 * p + 2];
    const float p0 = a00 * rx + a01 * ry + a02 * rz;   // A @ ray
    const float p1 = a10 * rx + a11 * ry + a12 * rz;
    const float p2 = a20 * rx + a21 * ry + a22 * rz;
    const float q0 = fmaf(dd[p], p0, b0);              // depth * (A@ray) + bv
    const float q1 = fmaf(dd[p], p1, b1);
    const float q2 = fmaf(dd[p], p2, b2);
    const float den = fmaxf(q2, 0.0f) + 1e-12f;        // relu(d) + eps
    const float inv = 1.0f / den;
    uv[2 * p + 0] = q0 * inv;
    uv[2 * p + 1] = q1 * inv;
    dv[p] = q2;
  }

  const v4f uvlo = {uv[0], uv[1], uv[2], uv[3]};
  const v4f uvhi = {uv[4], uv[5], uv[6], uv[7]};
  const v4f dvec = {dv[0], dv[1], dv[2], dv[3]};
  __builtin_nontemporal_store(uvlo, (v4f*)(out_uv + base * 2));
  __builtin_nontemporal_store(uvhi, (v4f*)(out_uv + base * 2 + 4));
  __builtin_nontemporal_store(dvec, (v4f*)(out_d + base));
}

// ---------------------------------------------------------------------------
extern "C" void kernel_launch(void* const* d_in, const int* in_sizes, int n_in,
                              void* d_out, int out_size, void* d_ws, size_t ws_size,
                              hipStream_t stream) {
  (void)n_in; (void)out_size; (void)ws_size;
  const float* depth0 = (const float*)d_in[0];  // (BS, H, W)
  const float* R0     = (const float*)d_in[1];  // (BS, 3, 3)
  const float* t0     = (const float*)d_in[2];  // (BS, 3)
  const float* R1     = (const float*)d_in[3];  // (BS, 3, 3)
  const float* t1     = (const float*)d_in[4];  // (BS, 3)
  const float* Km     = (const float*)d_in[5];  // (3, 3)
  const float* ray    = (const float*)d_in[6];  // (1, N, 3)

  const int npix = in_sizes[6] / 3;             // 262144
  const int bs   = in_sizes[0] / npix;          // 64

  float* out_uv = (float*)d_out;                              // (BS, N, 2)
  float* out_d  = (float*)d_out + (size_t)bs * npix * 2;      // (BS, N, 1)
  float* comp   = (float*)d_ws;                               // bs * 16 floats

  uvt_composite_kernel<<<dim3(bs), dim3(32), 0, stream>>>(R0, t0, R1, t1, Km, comp);

  const int pix_per_block = 256 * 4;
  dim3 grid((npix + pix_per_block - 1) / pix_per_block, bs);
  uvt_main_kernel<<<grid, dim3(256), 0, stream>>>(depth0, ray, comp, out_uv, out_d, npix);
}